// Differ_5703716569582
// MI455X (gfx1250) — compile-verified
//
#include <hip/hip_runtime.h>

// Problem size is fixed by the reference.
#define NDIM 4096
#define COLS_PER_BLOCK 1024   // 256 threads x 4 floats
#define ROWS_PER_BLOCK 8
#define THREADS 256

typedef __attribute__((ext_vector_type(4))) float v4f;
typedef __attribute__((ext_vector_type(4))) int   v4i;

// CDNA5 async global->LDS path (gfx1250). Guarded so the file compiles even
// if the toolchain lacks the builtin; fallback is plain b128 global loads.
#if defined(__gfx1250__) && __has_builtin(__builtin_amdgcn_global_load_async_to_lds_b128) && __has_builtin(__builtin_amdgcn_s_wait_asynccnt)
#define USE_ASYNC_LDS 1
// Builtin signature (from clang diagnostic): (v4i AS1*, v4i AS3*, imm i32 offset, imm i32 cpol)
#define ASYNC_B128(gsrc, ldst)                                                  \
  __builtin_amdgcn_global_load_async_to_lds_b128(                               \
      (__attribute__((address_space(1))) v4i*)(gsrc),                           \
      (__attribute__((address_space(3))) v4i*)(ldst), 0, 0)
#else
#define USE_ASYNC_LDS 0
#endif

// ---------------------------------------------------------------------------
// Kernel 1: extract diag(Sigma) into workspace (16 KB). Strided gather done
// exactly once so the streaming kernel only ever does coalesced loads.
// ---------------------------------------------------------------------------
__global__ void __launch_bounds__(THREADS) diag_extract_kernel(
    const float* __restrict__ Sigma, float* __restrict__ diag) {
  const int i = blockIdx.x * blockDim.x + threadIdx.x;
  if (i < NDIM) diag[i] = Sigma[(size_t)i * NDIM + i];
}

// ---------------------------------------------------------------------------
// Kernel 2: streaming elementwise kernel, bandwidth-bound (~200 MB total).
// Each thread owns 4 contiguous columns, loops over 8 rows. Sigma row chunks
// are double-buffered through LDS via async loads (one row in flight).
// Outputs use non-temporal b128 stores to keep Sigma resident in L2.
// ---------------------------------------------------------------------------
__global__ void __launch_bounds__(THREADS) differ_kernel(
    const float* __restrict__ mu, const float* __restrict__ Sigma,
    const float* __restrict__ diag, float* __restrict__ mu_d,
    float* __restrict__ sig_d) {
  const int t = threadIdx.x;
  const int k = blockIdx.x * COLS_PER_BLOCK + t * 4;  // this thread's columns
  const int row0 = blockIdx.y * ROWS_PER_BLOCK;

  // Column-constant values: live in registers, reused for all 8 rows.
  const v4f mu_k = *(const v4f*)(mu + k);
  const v4f d_k  = *(const v4f*)(diag + k);

#if USE_ASYNC_LDS
  __shared__ __align__(16) float sbuf[2][COLS_PER_BLOCK];
  // Prime the pipeline: row0's chunk -> sbuf[0].
  ASYNC_B128(Sigma + (size_t)row0 * NDIM + k, &sbuf[0][t * 4]);
#endif

#pragma unroll
  for (int r = 0; r < ROWS_PER_BLOCK; ++r) {
    const int row = row0 + r;
    const size_t idx = (size_t)row * NDIM + k;

#if USE_ASYNC_LDS
    if (r + 1 < ROWS_PER_BLOCK) {
      // Issue next row's async load, then wait until only it is in flight
      // (async loads complete in order within a wave -> row r is done).
      ASYNC_B128(Sigma + idx + NDIM, &sbuf[(r + 1) & 1][t * 4]);
      __builtin_amdgcn_s_wait_asynccnt(1);
    } else {
      __builtin_amdgcn_s_wait_asynccnt(0);
    }
    // Each lane reads exactly the 16 LDS bytes it requested: no barrier needed.
    const v4f S = *(const v4f*)(&sbuf[r & 1][t * 4]);
#else
    const v4f S = *(const v4f*)(Sigma + idx);
#endif

    // Uniform per-row scalars (scalar-cached loads).
    const float mu_j = mu[row];
    const float d_j  = diag[row];

    v4f md = mu_j - mu_k;                 // mu[j] - mu[k]
    v4f sd = (d_j + d_k) - 2.0f * S;      // S_jj + S_kk - 2*S_jk

    // Diagonal overwrite quirk: at j==k -> mu_d = -mu[j], sig_d = S_jj.
    const unsigned l = (unsigned)(row - k);
    if (l < 4u) {
      md[l] = -mu_j;
      sd[l] = d_j;
    }

    __builtin_nontemporal_store(md, (v4f*)(mu_d + idx));
    __builtin_nontemporal_store(sd, (v4f*)(sig_d + idx));
  }
}

// ---------------------------------------------------------------------------
// Launch: inputs are [mu (N), Sigma (N*N)] f32; output is [mu_d | sig_d],
// each N*N f32, concatenated. Workspace holds diag(Sigma) (N floats).
// ---------------------------------------------------------------------------
extern "C" void kernel_launch(void* const* d_in, const int* in_sizes, int n_in,
                              void* d_out, int out_size, void* d_ws, size_t ws_size,
                              hipStream_t stream) {
  (void)in_sizes; (void)n_in; (void)out_size; (void)ws_size;
  const float* mu    = (const float*)d_in[0];
  const float* Sigma = (const float*)d_in[1];
  float* out   = (float*)d_out;
  float* mu_d  = out;                           // N*N
  float* sig_d = out + (size_t)NDIM * NDIM;     // N*N
  float* diag  = (float*)d_ws;                  // N floats of scratch

  diag_extract_kernel<<<NDIM / THREADS, THREADS, 0, stream>>>(Sigma, diag);

  dim3 grid(NDIM / COLS_PER_BLOCK, NDIM / ROWS_PER_BLOCK);  // 4 x 512
  differ_kernel<<<grid, THREADS, 0, stream>>>(mu, Sigma, diag, mu_d, sig_d);
}